// Detect_72232759984313
// MI455X (gfx1250) — compile-verified
//
#include <hip/hip_runtime.h>
#include <stdint.h>

#define NUM_CLASSES 81
#define NUM_PRIORS  24564
#define TOPK        200
#define CONF_T      0.01f
#define NMS_T       0.45f
#define BLOCK       256

typedef __attribute__((address_space(1))) int gbl_int;   // global (AS1)
typedef __attribute__((address_space(3))) int lds_int;   // LDS (AS3)

// ---- CDNA5 async global->LDS copy (ASYNCcnt path) -------------------------
__device__ __forceinline__ void async_load_f32_to_lds(const float* gp, float* lp) {
#if defined(__gfx1250__)
#if __has_builtin(__builtin_amdgcn_global_load_async_to_lds_b32)
    __builtin_amdgcn_global_load_async_to_lds_b32((gbl_int*)gp, (lds_int*)lp, 0, 0);
#else
    unsigned lds_off = (unsigned)(uintptr_t)(lds_int*)lp;
    asm volatile("global_load_async_to_lds_b32 %0, %1, off"
                 :: "v"(lds_off), "v"(gp) : "memory");
#endif
#else
    *lp = *gp;  // host-pass / non-gfx1250 fallback
#endif
}

__device__ __forceinline__ void async_wait0() {
#if defined(__gfx1250__)
#if __has_builtin(__builtin_amdgcn_s_wait_asynccnt)
    __builtin_amdgcn_s_wait_asynccnt(0);
#else
    asm volatile("s_wait_asynccnt 0x0" ::: "memory");
#endif
#endif
}

__device__ __forceinline__ unsigned ballot32(bool p) {
#if defined(__gfx1250__) && __has_builtin(__builtin_amdgcn_ballot_w32)
    return __builtin_amdgcn_ballot_w32(p);
#else
    return (unsigned)__ballot(p);
#endif
}

// key = (float bits of score) : (~index)  -> max-order == (score desc, index asc)
__device__ __forceinline__ unsigned long long make_key(float v, int p) {
    return ((unsigned long long)__float_as_uint(v) << 32) | (unsigned)(~(unsigned)p);
}

// One block per (image b, class c>=1). Zero-fill of output done by zero_kernel.
__global__ __launch_bounds__(BLOCK)
void detect_kernel(const float* __restrict__ loc,     // [B,P,4]
                   const float* __restrict__ conf,    // [B,P,C]
                   const float* __restrict__ priors,  // [P,4]
                   float* __restrict__ out)           // [B,C,TOPK,5]
{
    __shared__ float               s_scores[NUM_PRIORS];   // 98256 B
    __shared__ unsigned            s_hist[256];
    __shared__ unsigned long long  s_key[256];
    __shared__ unsigned long long  s_thresh;
    __shared__ unsigned long long  s_prefix;
    __shared__ int                 s_shift;
    __shared__ unsigned            s_need;
    __shared__ unsigned            s_above;
    __shared__ int                 s_done;
    __shared__ unsigned            s_cnt;

    const int tid = threadIdx.x;
    const int c   = blockIdx.x + 1;     // class 0 stays zeroed
    const int b   = blockIdx.y;

    // ---------------- Phase A: async strided column -> LDS ----------------
    const float* col = conf + (size_t)b * NUM_PRIORS * NUM_CLASSES + c;
    for (int p = tid; p < NUM_PRIORS; p += BLOCK)
        async_load_f32_to_lds(col + (size_t)p * NUM_CLASSES, &s_scores[p]);
    async_wait0();
    __syncthreads();

    // ---------------- Phase B: 64-bit-key radix select (rank TOPK) --------
    if (tid == 0) {
        s_done = 0; s_need = TOPK; s_above = 0;
        s_prefix = 0ull; s_shift = 56; s_thresh = 0ull; s_cnt = 0;
    }
    __syncthreads();

    for (int pass = 0; pass < 8; ++pass) {
        if (s_done) break;                       // uniform (post-barrier)
        s_hist[tid] = 0u;
        __syncthreads();
        const int shift = s_shift;
        const unsigned long long prefix = s_prefix;
        for (int p = tid; p < NUM_PRIORS; p += BLOCK) {
            float v = s_scores[p];
            if (v > CONF_T) {
                unsigned long long k = make_key(v, p);
                if (shift == 56 || (k >> (shift + 8)) == prefix)
                    atomicAdd(&s_hist[(unsigned)((k >> shift) & 0xFFull)], 1u);
            }
        }
        __syncthreads();
        if (tid == 0) {
            if (pass == 0) {
                unsigned tot = 0;
                for (int d = 0; d < 256; ++d) tot += s_hist[d];
                if (tot <= TOPK) { s_thresh = 0ull; s_done = 1; }  // keep all candidates
            }
            if (!s_done) {
                unsigned need = s_need, cum = 0;
                int d;
                for (d = 255; d >= 0; --d) {
                    if (cum + s_hist[d] >= need) break;
                    cum += s_hist[d];
                }
                if (d < 0) d = 0;  // safety (cannot happen with consistent counts)
                s_above += cum;
                need    -= cum;
                unsigned long long npref = (s_prefix << 8) | (unsigned long long)(unsigned)d;
                if (s_hist[d] == need || s_shift == 0) {
                    s_thresh = npref << s_shift;   // inclusive threshold: #(key>=thr)==above+need
                    s_done   = 1;
                } else {
                    s_prefix = npref; s_need = need; s_shift -= 8;
                }
            }
        }
        __syncthreads();
    }

    // ---------------- Compaction of selected keys (<= 200) ----------------
    s_key[tid] = 0ull;
    __syncthreads();
    {
        const unsigned long long thr = s_thresh;
        for (int p = tid; p < NUM_PRIORS; p += BLOCK) {
            float v = s_scores[p];
            if (v > CONF_T) {
                unsigned long long k = make_key(v, p);
                if (k >= thr) {
                    unsigned slot = atomicAdd(&s_cnt, 1u);
                    if (slot < 256u) s_key[slot] = k;
                }
            }
        }
    }
    __syncthreads();

    // ---------------- Bitonic sort 256 keys, descending -------------------
    for (unsigned k = 2; k <= 256; k <<= 1) {
        for (unsigned j = k >> 1; j > 0; j >>= 1) {
            unsigned ixj = (unsigned)tid ^ j;
            if (ixj > (unsigned)tid) {
                unsigned long long a = s_key[tid], bb = s_key[ixj];
                bool up = (((unsigned)tid & k) == 0u);
                if ((a < bb) == up) { s_key[tid] = bb; s_key[ixj] = a; }
            }
            __syncthreads();
        }
    }

    // ---------------- Phase C/D: wave0 register-resident NMS ---------------
    if (tid < 32) {
        const int lane = tid;
        float x1r[7], y1r[7], x2r[7], y2r[7], arr[7], scr[7];
        unsigned validMask = 0, supMask = 0, keepMask = 0;

#pragma unroll
        for (int q = 0; q < 7; ++q) {
            int t = lane + 32 * q;
            float X1 = 0.f, Y1 = 0.f, X2 = 0.f, Y2 = 0.f, A = 0.f, S = 0.f;
            bool val = false;
            if (t < TOPK) {
                unsigned long long k = s_key[t];
                if (k != 0ull) {
                    val = true;
                    unsigned idx = ~((unsigned)(k & 0xFFFFFFFFull));
                    S = __uint_as_float((unsigned)(k >> 32));
                    const float* lp = loc + ((size_t)b * NUM_PRIORS + idx) * 4;
                    const float* pp = priors + (size_t)idx * 4;
                    float l0 = lp[0], l1 = lp[1], l2 = lp[2], l3 = lp[3];
                    float p0 = pp[0], p1 = pp[1], p2 = pp[2], p3 = pp[3];
                    float cx = p0 + l0 * 0.1f * p2;
                    float cy = p1 + l1 * 0.1f * p3;
                    float w  = p2 * expf(l2 * 0.2f);
                    float h  = p3 * expf(l3 * 0.2f);
                    X1 = cx - 0.5f * w; Y1 = cy - 0.5f * h;
                    X2 = cx + 0.5f * w; Y2 = cy + 0.5f * h;
                    A  = (X2 - X1) * (Y2 - Y1);
                }
            }
            x1r[q] = X1; y1r[q] = Y1; x2r[q] = X2; y2r[q] = Y2;
            arr[q] = A;  scr[q] = S;
            validMask |= (val ? 1u : 0u) << q;
        }

#pragma unroll
        for (int qi = 0; qi < 7; ++qi) {
            const int imax = (qi == 6) ? (TOPK - 6 * 32) : 32;
            for (int li = 0; li < imax; ++li) {
                const int i   = qi * 32 + li;
                float bx1 = __shfl(x1r[qi], li, 32);
                float by1 = __shfl(y1r[qi], li, 32);
                float bx2 = __shfl(x2r[qi], li, 32);
                float by2 = __shfl(y2r[qi], li, 32);
                float bar = __shfl(arr[qi], li, 32);
                unsigned vm = __shfl(validMask, li, 32);
                unsigned sm = __shfl(supMask,  li, 32);
                bool keep_i = ((vm >> qi) & 1u) && !((sm >> qi) & 1u);
                if (lane == li) keepMask |= (keep_i ? 1u : 0u) << qi;
                if (keep_i) {
#pragma unroll
                    for (int q = 0; q < 7; ++q) {
                        int j = lane + 32 * q;
                        float iw = fminf(bx2, x2r[q]) - fmaxf(bx1, x1r[q]);
                        float ih = fminf(by2, y2r[q]) - fmaxf(by1, y1r[q]);
                        iw = fmaxf(iw, 0.f); ih = fmaxf(ih, 0.f);
                        float inter = iw * ih;
                        float iou = inter / (bar + arr[q] - inter);
                        if (j > i && iou > NMS_T) supMask |= 1u << q;
                    }
                }
            }
        }

        // compact kept entries: pos = cumsum(keep)-1 across t order
        unsigned base = 0;
        float* oslice = out + ((size_t)b * NUM_CLASSES + c) * TOPK * 5;
#pragma unroll
        for (int q = 0; q < 7; ++q) {
            bool kp = (keepMask >> q) & 1u;
            unsigned mask = ballot32(kp);
            unsigned pos  = base + __popc(mask & ((1u << lane) - 1u));
            if (kp) {
                float* row = oslice + (size_t)pos * 5;
                row[0] = scr[q]; row[1] = x1r[q]; row[2] = y1r[q];
                row[3] = x2r[q]; row[4] = y2r[q];
            }
            base += __popc(mask);
        }
    }
}

__global__ void zero_kernel(float* __restrict__ out, int n) {
    int i = blockIdx.x * blockDim.x + threadIdx.x;
    int stride = gridDim.x * blockDim.x;
    for (; i < n; i += stride) out[i] = 0.f;
}

extern "C" void kernel_launch(void* const* d_in, const int* in_sizes, int n_in,
                              void* d_out, int out_size, void* d_ws, size_t ws_size,
                              hipStream_t stream) {
    (void)n_in; (void)d_ws; (void)ws_size;
    const float* loc    = (const float*)d_in[0];
    const float* conf   = (const float*)d_in[1];
    const float* priors = (const float*)d_in[2];
    float* out = (float*)d_out;

    const int B = in_sizes[0] / (NUM_PRIORS * 4);

    zero_kernel<<<1024, 256, 0, stream>>>(out, out_size);
    dim3 grid(NUM_CLASSES - 1, B);
    detect_kernel<<<grid, BLOCK, 0, stream>>>(loc, conf, priors, out);
}